// Panoformer_55310588837910
// MI455X (gfx1250) — compile-verified
//
#include <hip/hip_runtime.h>
#include <math.h>

#define Bb   4
#define Hh   128
#define Ww   256
#define Cc   32
#define Mm   2
#define Kpts 9
#define DKk  16
#define HIDn 128
#define TOK  (Bb*Hh*Ww)          // 131072
#define WAVES 4
#define TILE  16
#define BLK_TOK (WAVES*TILE)     // 64
#define NBLK (TOK/BLK_TOK)       // 2048

typedef float v2f __attribute__((ext_vector_type(2)));
typedef float v8f __attribute__((ext_vector_type(8)));

__device__ __forceinline__ float gelu_exact(float x) {
    return 0.5f * x * (1.0f + erff(x * 0.70710678118654752f));
}

__device__ __forceinline__ int iclamp(int v, int lo, int hi) {
    return v < lo ? lo : (v > hi ? hi : v);
}

// One wave computes a 16x16 f32 tile D = A(16xK) * B(Kx16) + C using
// V_WMMA_F32_16X16X4_F32, chaining K/4 accumulation steps.
// a: row-major [16][lda] (LDS), b: row-major [K][ldb] (global weights).
// r = lane&15, h2 = (lane>>4)*2 selects the K-pair per the ISA A/B layouts.
template <int K>
__device__ __forceinline__ v8f wmma_tile_f32(const float* __restrict__ a, int lda,
                                             const float* __restrict__ b, int ldb,
                                             v8f c, int r, int h2) {
#pragma unroll
    for (int kk = 0; kk < K; kk += 4) {
        v2f af, bf;
        af[0] = a[r * lda + kk + h2];
        af[1] = a[r * lda + kk + h2 + 1];
        bf[0] = b[(kk + h2) * ldb + r];
        bf[1] = b[(kk + h2 + 1) * ldb + r];
        c = __builtin_amdgcn_wmma_f32_16x16x4_f32(false, af, false, bf,
                                                  (short)0, c, false, false);
    }
    return c;
}

// ---------------------------------------------------------------------------
// Kernel 1: LN1 -> zq = y@wq, v = y@wk (WMMA); per-lane offsets/softmax heads.
// ---------------------------------------------------------------------------
__global__ __launch_bounds__(WAVES * 32)
void k_ln1_qkv(const float* __restrict__ x, const float* __restrict__ ref,
               const float* __restrict__ g1, const float* __restrict__ be1,
               const float* __restrict__ wq, const float* __restrict__ bq,
               const float* __restrict__ woff, const float* __restrict__ boff,
               const float* __restrict__ wA, const float* __restrict__ bA,
               const float* __restrict__ wk, const float* __restrict__ bk,
               float* __restrict__ v_out, float* __restrict__ A_out,
               float* __restrict__ pos_out) {
    __shared__ float Y[WAVES][TILE][33];
    __shared__ float ZQ[WAVES][TILE][33];
    const int w = threadIdx.x >> 5, lane = threadIdx.x & 31;
    const int r = lane & 15, hf = lane >> 4, h2 = hf * 2;
    const int t0 = (blockIdx.x * WAVES + w) * TILE;
    const int t  = t0 + r;

    // LayerNorm: lane (r,hf) owns channels [hf*16, hf*16+16) of token t.
    float xr[16];
    const float* xp = x + (size_t)t * Cc + hf * 16;
#pragma unroll
    for (int i = 0; i < 16; i += 4) {
        float4 q = *(const float4*)(xp + i);
        xr[i] = q.x; xr[i+1] = q.y; xr[i+2] = q.z; xr[i+3] = q.w;
    }
    float s = 0.f, s2 = 0.f;
#pragma unroll
    for (int i = 0; i < 16; ++i) { s += xr[i]; s2 += xr[i]*xr[i]; }
    s  += __shfl_xor(s, 16, 32);
    s2 += __shfl_xor(s2, 16, 32);
    const float mu  = s * (1.0f / Cc);
    const float var = s2 * (1.0f / Cc) - mu * mu;
    const float rs  = rsqrtf(var + 1e-5f);
#pragma unroll
    for (int i = 0; i < 16; ++i)
        Y[w][r][hf*16 + i] = (xr[i] - mu) * rs * g1[hf*16 + i] + be1[hf*16 + i];
    __syncthreads();

    // zq = Y @ wq + bq -> LDS
#pragma unroll
    for (int n = 0; n < 2; ++n) {
        v8f c = {};
        c = wmma_tile_f32<Cc>(&Y[w][0][0], 33, wq + n*16, Cc, c, r, h2);
#pragma unroll
        for (int i = 0; i < 8; ++i)
            ZQ[w][i + 8*hf][n*16 + r] = c[i] + bq[n*16 + r];
    }
    // v = Y @ wk + bk -> global
#pragma unroll
    for (int n = 0; n < 2; ++n) {
        v8f c = {};
        c = wmma_tile_f32<Cc>(&Y[w][0][0], 33, wk + n*16, Cc, c, r, h2);
#pragma unroll
        for (int i = 0; i < 8; ++i)
            v_out[(size_t)(t0 + i + 8*hf) * Cc + n*16 + r] = c[i] + bk[n*16 + r];
    }
    __syncthreads();

    // Per-lane: (row=r, head m=hf). 9 attn logits + softmax, 18 offsets -> pos.
    float zr[Cc];
#pragma unroll
    for (int i = 0; i < Cc; ++i) zr[i] = ZQ[w][r][i];
    const int m = hf;

    float lg[Kpts];
    float mx = -1e30f;
#pragma unroll
    for (int k = 0; k < Kpts; ++k) {
        float acc = bA[m*Kpts + k];
        for (int c = 0; c < Cc; ++c) acc += zr[c] * wA[c*(Mm*Kpts) + m*Kpts + k];
        lg[k] = acc; mx = fmaxf(mx, acc);
    }
    float se = 0.f;
#pragma unroll
    for (int k = 0; k < Kpts; ++k) { lg[k] = __expf(lg[k] - mx); se += lg[k]; }
    const float inv = 1.0f / se;
#pragma unroll
    for (int k = 0; k < Kpts; ++k)
        A_out[(size_t)t*(Mm*Kpts) + m*Kpts + k] = lg[k] * inv;

#pragma unroll
    for (int k = 0; k < Kpts; ++k) {
#pragma unroll
        for (int d = 0; d < 2; ++d) {
            const int o = m*(Kpts*2) + k*2 + d;
            float acc = boff[o];
            for (int c = 0; c < Cc; ++c) acc += zr[c] * woff[c*(2*Mm*Kpts) + o];
            pos_out[(size_t)t*(2*Mm*Kpts) + o] = acc + ref[(size_t)t*(Kpts*2) + k*2 + d];
        }
    }
}

// ---------------------------------------------------------------------------
// Kernel 2: bilinear gather + A-weighted sum -> @wm + bm + residual -> x1.
// ---------------------------------------------------------------------------
__global__ __launch_bounds__(WAVES * 32)
void k_attn(const float* __restrict__ x, const float* __restrict__ v,
            const float* __restrict__ A, const float* __restrict__ pos,
            const float* __restrict__ wm, const float* __restrict__ bm,
            float* __restrict__ x1) {
    __shared__ float OUT[WAVES][TILE][33];
    const int w = threadIdx.x >> 5, lane = threadIdx.x & 31;
    const int r = lane & 15, hf = lane >> 4, h2 = hf * 2;
    const int t0 = (blockIdx.x * WAVES + w) * TILE;
    const int t  = t0 + r;
    const int b  = t >> 15;            // / (H*W)
    const int m  = hf;
    const float* vb = v + (size_t)b * (Hh*Ww*Cc) + m * DKk;

    float acc[DKk];
#pragma unroll
    for (int i = 0; i < DKk; ++i) acc[i] = 0.f;

    for (int k = 0; k < Kpts; ++k) {
        const float a  = A[(size_t)t*(Mm*Kpts) + m*Kpts + k];
        const float px = pos[(size_t)t*(2*Mm*Kpts) + m*(Kpts*2) + k*2 + 0];
        const float py = pos[(size_t)t*(2*Mm*Kpts) + m*(Kpts*2) + k*2 + 1];
        const float xf = (px + 1.0f) * 0.5f * (float)(Ww - 1);
        const float yf = (py + 1.0f) * 0.5f * (float)(Hh - 1);
        const float x0f = floorf(xf), y0f = floorf(yf);
        const float wx = xf - x0f, wy = yf - y0f;
        const int x0 = iclamp((int)x0f,     0, Ww - 1);
        const int x1c= iclamp((int)x0f + 1, 0, Ww - 1);
        const int y0 = iclamp((int)y0f,     0, Hh - 1);
        const int y1 = iclamp((int)y0f + 1, 0, Hh - 1);
        const float* p00 = vb + ((size_t)y0*Ww + x0 ) * Cc;
        const float* p01 = vb + ((size_t)y0*Ww + x1c) * Cc;
        const float* p10 = vb + ((size_t)y1*Ww + x0 ) * Cc;
        const float* p11 = vb + ((size_t)y1*Ww + x1c) * Cc;
        const float w00 = (1.f-wx)*(1.f-wy)*a, w01 = wx*(1.f-wy)*a;
        const float w10 = (1.f-wx)*wy*a,       w11 = wx*wy*a;
#pragma unroll
        for (int i = 0; i < DKk; i += 4) {
            float4 q00 = *(const float4*)(p00 + i);
            float4 q01 = *(const float4*)(p01 + i);
            float4 q10 = *(const float4*)(p10 + i);
            float4 q11 = *(const float4*)(p11 + i);
            acc[i+0] += w00*q00.x + w01*q01.x + w10*q10.x + w11*q11.x;
            acc[i+1] += w00*q00.y + w01*q01.y + w10*q10.y + w11*q11.y;
            acc[i+2] += w00*q00.z + w01*q01.z + w10*q10.z + w11*q11.z;
            acc[i+3] += w00*q00.w + w01*q01.w + w10*q10.w + w11*q11.w;
        }
    }
#pragma unroll
    for (int i = 0; i < DKk; ++i) OUT[w][r][m*DKk + i] = acc[i];
    __syncthreads();

#pragma unroll
    for (int n = 0; n < 2; ++n) {
        v8f c = {};
        c = wmma_tile_f32<Cc>(&OUT[w][0][0], 33, wm + n*16, Cc, c, r, h2);
#pragma unroll
        for (int i = 0; i < 8; ++i) {
            const int row = i + 8*hf, col = n*16 + r;
            const size_t idx = (size_t)(t0 + row) * Cc + col;
            x1[idx] = x[idx] + c[i] + bm[col];
        }
    }
}

// ---------------------------------------------------------------------------
// Kernel 3: LN2 -> h = GELU(t @ w1 + b1)   (32 -> 128, 64 WMMAs per wave)
// ---------------------------------------------------------------------------
__global__ __launch_bounds__(WAVES * 32)
void k_ln2_w1(const float* __restrict__ x1, const float* __restrict__ g2,
              const float* __restrict__ be2, const float* __restrict__ w1,
              const float* __restrict__ b1, float* __restrict__ hbuf) {
    __shared__ float T[WAVES][TILE][33];
    const int w = threadIdx.x >> 5, lane = threadIdx.x & 31;
    const int r = lane & 15, hf = lane >> 4, h2 = hf * 2;
    const int t0 = (blockIdx.x * WAVES + w) * TILE;
    const int t  = t0 + r;

    float xr[16];
    const float* xp = x1 + (size_t)t * Cc + hf * 16;
#pragma unroll
    for (int i = 0; i < 16; i += 4) {
        float4 q = *(const float4*)(xp + i);
        xr[i] = q.x; xr[i+1] = q.y; xr[i+2] = q.z; xr[i+3] = q.w;
    }
    float s = 0.f, s2 = 0.f;
#pragma unroll
    for (int i = 0; i < 16; ++i) { s += xr[i]; s2 += xr[i]*xr[i]; }
    s  += __shfl_xor(s, 16, 32);
    s2 += __shfl_xor(s2, 16, 32);
    const float mu  = s * (1.0f / Cc);
    const float var = s2 * (1.0f / Cc) - mu * mu;
    const float rs  = rsqrtf(var + 1e-5f);
#pragma unroll
    for (int i = 0; i < 16; ++i)
        T[w][r][hf*16 + i] = (xr[i] - mu) * rs * g2[hf*16 + i] + be2[hf*16 + i];
    __syncthreads();

#pragma unroll
    for (int n = 0; n < 8; ++n) {
        v8f c = {};
        c = wmma_tile_f32<Cc>(&T[w][0][0], 33, w1 + n*16, HIDn, c, r, h2);
#pragma unroll
        for (int i = 0; i < 8; ++i) {
            const int row = i + 8*hf, col = n*16 + r;
            hbuf[(size_t)(t0 + row) * HIDn + col] = gelu_exact(c[i] + b1[col]);
        }
    }
}

// ---------------------------------------------------------------------------
// Kernel 4: depthwise 3x3 (circular W, zero H) + bdw + GELU -> @w2 + b2 + res.
// ---------------------------------------------------------------------------
__global__ __launch_bounds__(WAVES * 32)
void k_leff_out(const float* __restrict__ hbuf, const float* __restrict__ wdw,
                const float* __restrict__ bdw, const float* __restrict__ w2,
                const float* __restrict__ b2, float* __restrict__ xio) {
    __shared__ float G[WAVES][TILE][HIDn + 1];
    const int w = threadIdx.x >> 5, lane = threadIdx.x & 31;
    const int r = lane & 15, hf = lane >> 4, h2 = hf * 2;
    const int t0 = (blockIdx.x * WAVES + w) * TILE;
    const int t  = t0 + r;
    const int b  = t >> 15;
    const int rem = t & 32767;
    const int hh = rem >> 8;       // row in [0,128)
    const int ww = rem & 255;      // col in [0,256)

    const float* np[9];
#pragma unroll
    for (int dy = 0; dy < 3; ++dy) {
        const int hy = hh + dy - 1;
        const bool ok = (hy >= 0) && (hy < Hh);
#pragma unroll
        for (int dx = 0; dx < 3; ++dx) {
            const int wxp = (ww + dx - 1 + Ww) & (Ww - 1);   // circular width
            np[dy*3 + dx] = ok ? hbuf + (size_t)(((b*Hh + hy)*Ww) + wxp) * HIDn
                               : (const float*)0;            // zero height pad
        }
    }

    // Lane covers 64 channels (hf*64..), in 4 chunks of 16.
#pragma unroll
    for (int cb = 0; cb < 4; ++cb) {
        const int cbase = hf*64 + cb*16;
        float acc[16];
#pragma unroll
        for (int i = 0; i < 16; ++i) acc[i] = bdw[cbase + i];
#pragma unroll
        for (int tap = 0; tap < 9; ++tap) {
            const float* p = np[tap];
            if (p) {
#pragma unroll
                for (int i = 0; i < 16; i += 4) {
                    float4 q = *(const float4*)(p + cbase + i);
                    acc[i+0] += q.x * wdw[(cbase+i+0)*9 + tap];
                    acc[i+1] += q.y * wdw[(cbase+i+1)*9 + tap];
                    acc[i+2] += q.z * wdw[(cbase+i+2)*9 + tap];
                    acc[i+3] += q.w * wdw[(cbase+i+3)*9 + tap];
                }
            }
        }
#pragma unroll
        for (int i = 0; i < 16; ++i) G[w][r][cbase + i] = gelu_exact(acc[i]);
    }
    __syncthreads();

#pragma unroll
    for (int n = 0; n < 2; ++n) {
        v8f c = {};
        c = wmma_tile_f32<HIDn>(&G[w][0][0], HIDn + 1, w2 + n*16, Cc, c, r, h2);
#pragma unroll
        for (int i = 0; i < 8; ++i) {
            const int row = i + 8*hf, col = n*16 + r;
            const size_t idx = (size_t)(t0 + row) * Cc + col;
            xio[idx] = xio[idx] + c[i] + b2[col];
        }
    }
}

// ---------------------------------------------------------------------------
extern "C" void kernel_launch(void* const* d_in, const int* in_sizes, int n_in,
                              void* d_out, int out_size, void* d_ws, size_t ws_size,
                              hipStream_t stream) {
    const float* x    = (const float*)d_in[0];
    const float* ref  = (const float*)d_in[1];
    const float* g1   = (const float*)d_in[2];
    const float* be1  = (const float*)d_in[3];
    const float* wq   = (const float*)d_in[4];
    const float* bq   = (const float*)d_in[5];
    const float* woff = (const float*)d_in[6];
    const float* boff = (const float*)d_in[7];
    const float* wA   = (const float*)d_in[8];
    const float* bA   = (const float*)d_in[9];
    const float* wk   = (const float*)d_in[10];
    const float* bk   = (const float*)d_in[11];
    const float* wm   = (const float*)d_in[12];
    const float* bm   = (const float*)d_in[13];
    const float* g2   = (const float*)d_in[14];
    const float* be2  = (const float*)d_in[15];
    const float* w1   = (const float*)d_in[16];
    const float* b1   = (const float*)d_in[17];
    const float* wdw  = (const float*)d_in[18];
    const float* bdw  = (const float*)d_in[19];
    const float* w2   = (const float*)d_in[20];
    const float* b2   = (const float*)d_in[21];
    float* out = (float*)d_out;

    float* ws = (float*)d_ws;
    size_t off = 0;
    float* vbuf = ws + off; off += (size_t)TOK * Cc;              // 16.8 MB
    float* Abuf = ws + off; off += (size_t)TOK * Mm * Kpts;       //  9.4 MB
    float* pbuf = ws + off; off += (size_t)TOK * 2 * Mm * Kpts;   // 18.9 MB
    float* hbuf = ws + off; off += (size_t)TOK * HIDn;            // 67.1 MB

    dim3 blk(WAVES * 32);
    dim3 grd(NBLK);
    k_ln1_qkv<<<grd, blk, 0, stream>>>(x, ref, g1, be1, wq, bq, woff, boff,
                                       wA, bA, wk, bk, vbuf, Abuf, pbuf);
    k_attn<<<grd, blk, 0, stream>>>(x, vbuf, Abuf, pbuf, wm, bm, out);
    k_ln2_w1<<<grd, blk, 0, stream>>>(out, g2, be2, w1, b1, hbuf);
    k_leff_out<<<grd, blk, 0, stream>>>(hbuf, wdw, bdw, w2, b2, out);
}